// AutoEncoderLoss_11656541241933
// MI455X (gfx1250) — compile-verified
//
#include <hip/hip_runtime.h>

typedef __attribute__((ext_vector_type(2))) float v2f;
typedef __attribute__((ext_vector_type(8))) float v8f;

// One wave (32 threads) per block. Each wave owns TWO tiles of 16 "row"
// points (32 rows total) from P in one batch and scans all nQ "column"
// points from Q in tiles of 16 via V_WMMA_F32_16X16X4_F32:
//   A row i  = [x0, x1, x2, 1]        (loop-invariant, 2 VGPRs)
//   B col j  = [-2y0, -2y1, -2y2, yy] (rebuilt per tile, shared by 2 WMMAs)
//   C        = inline 0
//   D[i][j]  = -2 x_i . y_j + |y_j|^2
// The row-constant |x_i|^2 commutes with min_j and is added after reduction.
__global__ __launch_bounds__(32)
void chamfer_dir_kernel(const float* __restrict__ P,
                        const float* __restrict__ Q,
                        float* __restrict__ partial,
                        int nP, int nQ)
{
    const int lane = threadIdx.x;      // 0..31, wave32
    const int col  = lane & 15;        // point-in-tile index
    const bool hi  = lane >= 16;

    const int groupsPerBatch = nP >> 5;              // 32 rows per block
    const int b       = blockIdx.x / groupsPerBatch;
    const int rowBase = (blockIdx.x % groupsPerBatch) << 5;

    // ---- Two loop-invariant A tiles: rows [rowBase..+15], [rowBase+16..+31]
    // ISA 32-bit 16x4 A layout: lanes 0-15 -> (K=0,K=1), lanes 16-31 -> (K=2,K=3)
    const float* pb0 = P + ((size_t)b * nP + (size_t)(rowBase + col)) * 3;
    const float* pb1 = pb0 + 48;                     // +16 points * 3 floats
    const float p0x = pb0[0], p0y = pb0[1], p0z = pb0[2];
    const float p1x = pb1[0], p1y = pb1[1], p1z = pb1[2];
    const float pp0 = p0x * p0x + p0y * p0y + p0z * p0z;
    const float pp1 = p1x * p1x + p1y * p1y + p1z * p1z;
    v2f a0, a1;
    a0.x = hi ? p0z : p0x;  a0.y = hi ? 1.0f : p0y;
    a1.x = hi ? p1z : p1x;  a1.y = hi ? 1.0f : p1y;

    // Running elementwise mins of D tiles (C/D 16x16 f32 layout).
    v8f mn0, mn1;
#pragma unroll
    for (int r = 0; r < 8; ++r) { mn0[r] = 3.4e38f; mn1[r] = 3.4e38f; }

    const v8f zeroC = {};   // lowers to inline SRC2 = 0

    const float* qb = Q + (size_t)b * nQ * 3;
    for (int t = 0; t < nQ; t += 16) {
        // ---- B tile: column j = t + col, col = [-2y0, -2y1, -2y2, |y|^2] --
        const float* qp = qb + (size_t)(t + col) * 3;
        const float qx = qp[0], qy = qp[1], qz = qp[2];
        const float qq = qx * qx + qy * qy + qz * qz;
        v2f bb;
        bb.x = hi ? (-2.0f * qz) : (-2.0f * qx);
        bb.y = hi ? qq           : (-2.0f * qy);

        v8f d0 = __builtin_amdgcn_wmma_f32_16x16x4_f32(
            false, a0, false, bb, (short)0, zeroC, false, false);
        v8f d1 = __builtin_amdgcn_wmma_f32_16x16x4_f32(
            false, a1, false, bb, (short)0, zeroC, false, false);

#pragma unroll
        for (int r = 0; r < 8; ++r) {
            mn0[r] = fminf(mn0[r], d0[r]);
            mn1[r] = fminf(mn1[r], d1[r]);
        }
    }

    // ---- Row-min across the 16 lanes of each half-wave --------------------
    // VGPR r: lanes 0-15 hold row M=r, lanes 16-31 hold row M=r+8.
#pragma unroll
    for (int r = 0; r < 8; ++r) {
        float v0 = mn0[r], v1 = mn1[r];
        v0 = fminf(v0, __shfl_xor(v0, 1, 32));
        v1 = fminf(v1, __shfl_xor(v1, 1, 32));
        v0 = fminf(v0, __shfl_xor(v0, 2, 32));
        v1 = fminf(v1, __shfl_xor(v1, 2, 32));
        v0 = fminf(v0, __shfl_xor(v0, 4, 32));
        v1 = fminf(v1, __shfl_xor(v1, 4, 32));
        v0 = fminf(v0, __shfl_xor(v0, 8, 32));
        v1 = fminf(v1, __shfl_xor(v1, 8, 32));
        mn0[r] = v0; mn1[r] = v1;
    }
    float s = 0.0f;
#pragma unroll
    for (int r = 0; r < 8; ++r) s += mn0[r] + mn1[r];
    s += __shfl_xor(s, 16, 32);           // rows 0-7 halves + rows 8-15 halves

    // ---- Add the deferred sum of |x_i|^2 over all 32 rows -----------------
    float xs = pp0 + pp1;                 // both halves hold identical copies
    xs += __shfl_xor(xs, 1, 32);
    xs += __shfl_xor(xs, 2, 32);
    xs += __shfl_xor(xs, 4, 32);
    xs += __shfl_xor(xs, 8, 32);          // lanes 0-15: sum over 32 points

    if (lane == 0) partial[blockIdx.x] = s + xs;
}

// Deterministic final reduction (fixed order, no float atomics).
__global__ __launch_bounds__(256)
void chamfer_reduce_kernel(const float* __restrict__ partial,
                           float* __restrict__ out, int n)
{
    __shared__ float sm[256];
    float s = 0.0f;
    for (int i = threadIdx.x; i < n; i += 256) s += partial[i];
    sm[threadIdx.x] = s;
    __syncthreads();
    for (int w = 128; w > 0; w >>= 1) {
        if ((int)threadIdx.x < w) sm[threadIdx.x] += sm[threadIdx.x + w];
        __syncthreads();
    }
    if (threadIdx.x == 0) out[0] = sm[0];
}

extern "C" void kernel_launch(void* const* d_in, const int* in_sizes, int n_in,
                              void* d_out, int out_size, void* d_ws, size_t ws_size,
                              hipStream_t stream) {
    const float* pred = (const float*)d_in[0];   // [B, N, 3] f32
    const float* targ = (const float*)d_in[1];   // [B, M, 3] f32
    float* out = (float*)d_out;                  // scalar f32
    float* ws  = (float*)d_ws;                   // per-block partials

    const int B = 16, D = 3;
    const int N = in_sizes[0] / (B * D);
    const int M = in_sizes[1] / (B * D);

    const int blocks0 = B * (N >> 5);   // 32 prediction rows per block
    const int blocks1 = B * (M >> 5);   // 32 target rows per block

    chamfer_dir_kernel<<<blocks0, 32, 0, stream>>>(pred, targ, ws, N, M);
    chamfer_dir_kernel<<<blocks1, 32, 0, stream>>>(targ, pred, ws + blocks0, M, N);
    chamfer_reduce_kernel<<<1, 256, 0, stream>>>(ws, out, blocks0 + blocks1);
}